// LearnableKalmanTracker_61048665145763
// MI455X (gfx1250) — compile-verified
//
#include <hip/hip_runtime.h>
#include <hip/hip_bf16.h>
#include <math.h>

// LearnableKalmanTracker on gfx1250 (MI455X).
// Latency-bound serial scan: 2 batches per wave, 8x8 covariances packed
// block-diagonally into 16x16 tiles, propagated with V_WMMA_F32_16X16X4_F32.
// All state lives in LDS/registers; HBM traffic is z-loads + 4-float stores.
// Round 2: fragment loads hoisted into register arrays so the compiler can
// clause DS loads and overlap their latency with WMMA issue; float4
// measurement ingest (1 vmem + 1 ds op per batch per step).

typedef __attribute__((ext_vector_type(2))) float v2f;
typedef __attribute__((ext_vector_type(8))) float v8f;

#define WPB 8          // waves per block (256 threads, wave32)
#define TS 17          // LDS tile row stride (pad to dodge bank conflicts)

static __device__ __forceinline__ v8f wmma4(v2f a, v2f b, v8f c) {
  // D(16x16) = A(16x4) * B(4x16) + C, fp32.
  return __builtin_amdgcn_wmma_f32_16x16x4_f32(false, a, false, b, (short)0, c,
                                               false, false);
}

// sVec layout (per wave, floats): mean(16) pm(16) innov(8) z(8) S(32) Sinv(32)
#define O_MEAN 0
#define O_PM   16
#define O_INN  32
#define O_Z    40
#define O_SM   48
#define O_SI   80

__global__ __launch_bounds__(256)
void kalman_scan_kernel(const float* __restrict__ meas,
                        const float* __restrict__ F_diag,
                        const float* __restrict__ F_vel,
                        const float* __restrict__ Hb,
                        const float* __restrict__ q_scale,
                        const float* __restrict__ r_scale,
                        const float* __restrict__ mwp,
                        const float* __restrict__ W1,
                        const float* __restrict__ b1,
                        const float* __restrict__ g1,
                        const float* __restrict__ be1,
                        const float* __restrict__ W2,
                        const float* __restrict__ b2,
                        float* __restrict__ out,
                        int B, int T)
{
  __shared__ float sF[8][8];
  __shared__ float sH[4][8];
  __shared__ float sW1[16][8];
  __shared__ float sb1[16], sg1[16], sbe1[16], sW2v[16];
  __shared__ float sScal[4];
  __shared__ float sCov[WPB][16][TS];   // persistent block-diag covariance
  __shared__ float sU  [WPB][16][TS];   // F*cov scratch
  __shared__ float sP  [WPB][16][TS];   // predicted covariance
  __shared__ float sK  [WPB][16][9];    // -K, 16x8 blocked
  __shared__ float sHP [WPB][8][TS];    // H*P, 8x16 blocked
  __shared__ float sVec[WPB][128];

  const int tid = threadIdx.x;

  // ---- block-wide constant staging (once) ----
  if (tid < 64) {
    int i = tid >> 3, j = tid & 7;
    float v = 0.f;
    if (i == j) v = fminf(fmaxf(F_diag[i], 0.9f), 1.1f);
    if (i < 4 && j == i + 4) v = 1.f / (1.f + __expf(-F_vel[i]));
    sF[i][j] = v;
  }
  if (tid < 32)  sH[tid >> 3][tid & 7] = Hb[tid];
  if (tid < 128) sW1[tid >> 3][tid & 7] = W1[tid];
  if (tid < 16) { sb1[tid] = b1[tid]; sg1[tid] = g1[tid];
                  sbe1[tid] = be1[tid]; sW2v[tid] = W2[tid]; }
  if (tid == 0) {
    sScal[0] = __expf(q_scale[0]);
    sScal[1] = __expf(r_scale[0]);
    sScal[2] = 1.f / (1.f + __expf(-mwp[0]));
    sScal[3] = b2[0];
  }
  __syncthreads();

  const int lane = tid & 31;
  const int w    = tid >> 5;
  const int gw   = blockIdx.x * WPB + w;
  const int b0   = gw * 2;
  if (b0 >= B) return;                 // no barriers after this point

  const int bat = lane >> 4;           // which batch (0/1) this lane serves
  const int li  = lane & 15;
  const int hi  = bat;                 // upper half-wave flag
  const int bo  = bat * 8;             // block offset inside 16x16 tiles
  const long bb = (long)(b0 + bat);
  const long rowStride = (long)T * 4;

  const float qv = sScal[0], rv = sScal[1], sigmw = sScal[2], b2v = sScal[3];

  // ---- precompute A-fragments of F_blk = diag(F,F), 4 K-slices ----
  // A layout (16x4 f32): lanes 0-15 -> M=lane, K=0(V0)/1(V1);
  //                      lanes 16-31 -> M=lane-16, K=2/3.
  v2f Fa[4];
#pragma unroll
  for (int k0 = 0; k0 < 4; ++k0) {
    int K = 4 * k0 + 2 * hi;
    float x = 0.f, y = 0.f;
    if (li < 8) { if (K < 8)  x = sF[li][K];       if (K + 1 < 8)  y = sF[li][K + 1]; }
    else        { if (K >= 8) x = sF[li - 8][K - 8]; if (K + 1 >= 8) y = sF[li - 8][K - 7]; }
    v2f f; f.x = x; f.y = y; Fa[k0] = f;
  }

  // ---- per-wave state init (t = 0) ----
  for (int idx = lane; idx < 256; idx += 32) {
    int r = idx >> 4, c = idx & 15;
    sCov[w][r][c] = (r == c) ? 1.f : 0.f;        // cov0 = I8 per batch
  }
  for (int idx = lane; idx < 16 * 9; idx += 32) sK[w][idx / 9][idx % 9] = 0.f;
  for (int idx = lane; idx < 8 * TS; idx += 32) sHP[w][idx / TS][idx % TS] = 0.f;
  if (li == 0) {
    float4 z0 = *(const float4*)(meas + bb * rowStride);
    float4 zr; zr.x = 0.f; zr.y = 0.f; zr.z = 0.f; zr.w = 0.f;
    *(float4*)&sVec[w][O_MEAN + bo]     = z0;    // mean0[:4] = z0
    *(float4*)&sVec[w][O_MEAN + bo + 4] = zr;    // mean0[4:] = 0
    *(float4*)(out + bb * rowStride)    = z0;    // out[:,0,:] = z0
  }

  // ---- serial scan over time ----
  for (int t = 1; t < T; ++t) {
    // measurement ingest: 1 x b128 load + 1 x b128 LDS store per batch
    if (li == 0) {
      const float* zp = meas + bb * rowStride + (long)t * 4;
      float4 zv = *(const float4*)zp;
      *(float4*)&sVec[w][O_Z + bat * 4] = zv;
      if (t + 1 < T) __builtin_prefetch(zp + 4, 0, 1);
    }

    // Phase A: U = F_blk * cov
    // B frag slice k0: V0=(row K, col li), V1=(row K+1, col li)
    v2f bA[4];
#pragma unroll
    for (int k0 = 0; k0 < 4; ++k0) {
      int K = 4 * k0 + 2 * hi;
      bA[k0].x = sCov[w][K][li]; bA[k0].y = sCov[w][K + 1][li];
    }
    v8f u = {};
#pragma unroll
    for (int k0 = 0; k0 < 4; ++k0) u = wmma4(Fa[k0], bA[k0], u);
#pragma unroll
    for (int r = 0; r < 8; ++r) sU[w][r + 8 * hi][li] = u[r];

    // Phase B: P = F_blk * U^T + q*I  (cov symmetric => F cov F^T = F (F cov)^T)
    v2f bB[4];
#pragma unroll
    for (int k0 = 0; k0 < 4; ++k0) {
      int K = 4 * k0 + 2 * hi;
      bB[k0].x = sU[w][li][K]; bB[k0].y = sU[w][li][K + 1];  // U^T(K,N)=U(N,K)
    }
    v8f p;
#pragma unroll
    for (int r = 0; r < 8; ++r) p[r] = ((r + 8 * hi) == li) ? qv : 0.f;
#pragma unroll
    for (int k0 = 0; k0 < 4; ++k0) p = wmma4(Fa[k0], bB[k0], p);
#pragma unroll
    for (int r = 0; r < 8; ++r) sP[w][r + 8 * hi][li] = p[r];

    // ---- Phase C: scalar lane work (DS ops of one wave are in-order) ----
    // pm = F * mean
    if (li < 8) {
      float a = 0.f;
#pragma unroll
      for (int j = 0; j < 8; ++j) a += sF[li][j] * sVec[w][O_MEAN + bo + j];
      sVec[w][O_PM + bo + li] = a;
    }
    // innov = z - H*pm
    if (li < 4) {
      float a = sVec[w][O_Z + bat * 4 + li];
#pragma unroll
      for (int j = 0; j < 8; ++j) a -= sH[li][j] * sVec[w][O_PM + bo + j];
      sVec[w][O_INN + bat * 4 + li] = a;
    }
    // HP = H * P (4x8 per batch) -> blocked 8x16 (off-blocks stay zero)
#pragma unroll
    for (int rep = 0; rep < 2; ++rep) {
      int e = li + 16 * rep;
      int i = e >> 3, c = e & 7;
      float a = 0.f;
#pragma unroll
      for (int k = 0; k < 8; ++k) a += sH[i][k] * sP[w][bo + k][bo + c];
      sHP[w][bat * 4 + i][bo + c] = a;
    }
    // S = HP * H^T + r*I  (16 entries per batch, one per lane)
    {
      int i = li >> 2, jj = li & 3;
      float a = (i == jj) ? rv : 0.f;
#pragma unroll
      for (int c = 0; c < 8; ++c) a += sHP[w][bat * 4 + i][bo + c] * sH[jj][c];
      sVec[w][O_SM + bat * 16 + li] = a;
    }
    // Sinv: one lane per batch, 4x4 adjugate inverse
    if (li == 0) {
      const float* sm = &sVec[w][O_SM + bat * 16];
      float*       si = &sVec[w][O_SI + bat * 16];
      float a00=sm[0],a01=sm[1],a02=sm[2],a03=sm[3];
      float a10=sm[4],a11=sm[5],a12=sm[6],a13=sm[7];
      float a20=sm[8],a21=sm[9],a22=sm[10],a23=sm[11];
      float a30=sm[12],a31=sm[13],a32=sm[14],a33=sm[15];
      float c0=a00*a11-a01*a10, c1=a00*a12-a02*a10, c2=a00*a13-a03*a10;
      float c3=a01*a12-a02*a11, c4=a01*a13-a03*a11, c5=a02*a13-a03*a12;
      float d0=a20*a31-a21*a30, d1=a20*a32-a22*a30, d2=a20*a33-a23*a30;
      float d3=a21*a32-a22*a31, d4=a21*a33-a23*a31, d5=a22*a33-a23*a32;
      float det = c0*d5 - c1*d4 + c2*d3 + c3*d2 - c4*d1 + c5*d0;
      float id = 1.f / det;
      si[0] =( a11*d5 - a12*d4 + a13*d3)*id;
      si[1] =(-a01*d5 + a02*d4 - a03*d3)*id;
      si[2] =( a31*c5 - a32*c4 + a33*c3)*id;
      si[3] =(-a21*c5 + a22*c4 - a23*c3)*id;
      si[4] =(-a10*d5 + a12*d2 - a13*d1)*id;
      si[5] =( a00*d5 - a02*d2 + a03*d1)*id;
      si[6] =(-a30*c5 + a32*c2 - a33*c1)*id;
      si[7] =( a20*c5 - a22*c2 + a23*c1)*id;
      si[8] =( a10*d4 - a11*d2 + a13*d0)*id;
      si[9] =(-a00*d4 + a01*d2 - a03*d0)*id;
      si[10]=( a30*c4 - a31*c2 + a33*c0)*id;
      si[11]=(-a20*c4 + a21*c2 - a23*c0)*id;
      si[12]=(-a10*d3 + a11*d1 - a12*d0)*id;
      si[13]=( a00*d3 - a01*d1 + a02*d0)*id;
      si[14]=(-a30*c3 + a31*c1 - a32*c0)*id;
      si[15]=( a20*c3 - a21*c1 + a22*c0)*id;
    }
    // K = P H^T Sinv = (HP)^T Sinv  (P symmetric); store -K blocked 16x8
#pragma unroll
    for (int rep = 0; rep < 2; ++rep) {
      int e = li + 16 * rep;
      int a = e >> 2, i = e & 3;
      float acc = 0.f;
#pragma unroll
      for (int j = 0; j < 4; ++j)
        acc += sHP[w][bat * 4 + j][bo + a] * sVec[w][O_SI + bat * 16 + 4 * j + i];
      sK[w][bo + a][bat * 4 + i] = -acc;
    }

    // Phase E fragment loads hoisted here: they overlap the MLP (pure
    // VALU/shfl) below. DS ops of one wave are in-order, so these see the
    // sK/sHP stores above.
    v2f aE[2], bE[2];
#pragma unroll
    for (int k0 = 0; k0 < 2; ++k0) {
      int K = 4 * k0 + 2 * hi;
      aE[k0].x = sK[w][li][K];  aE[k0].y = sK[w][li][K + 1];
      bE[k0].x = sHP[w][K][li]; bE[k0].y = sHP[w][K + 1][li];
    }

    // MLP confidence: lane li computes hidden unit li for its batch
    float pre = sb1[li];
#pragma unroll
    for (int i2 = 0; i2 < 4; ++i2) pre += sW1[li][i2]     * sVec[w][O_INN + bat * 4 + i2];
#pragma unroll
    for (int i2 = 0; i2 < 4; ++i2) pre += sW1[li][4 + i2] * sVec[w][O_Z   + bat * 4 + i2];
    float s = pre;
#pragma unroll
    for (int m = 8; m >= 1; m >>= 1) s += __shfl_xor(s, m, 32);
    float mu = s * 0.0625f;
    float d  = pre - mu;
    float vs = d * d;
#pragma unroll
    for (int m = 8; m >= 1; m >>= 1) vs += __shfl_xor(vs, m, 32);
    float hval = d * rsqrtf(vs * 0.0625f + 1e-5f) * sg1[li] + sbe1[li];
    hval = fmaxf(hval, 0.f);
    float cs = hval * sW2v[li];
#pragma unroll
    for (int m = 8; m >= 1; m >>= 1) cs += __shfl_xor(cs, m, 32);
    float conf = 1.f / (1.f + __expf(-(cs + b2v)));
    float wgt = conf * sigmw;

    // new_mean = pm + w * K*innov  (K = -negK) ; emit output row
    if (li < 8) {
      float kin = 0.f;
#pragma unroll
      for (int i2 = 0; i2 < 4; ++i2)
        kin += sK[w][bo + li][bat * 4 + i2] * sVec[w][O_INN + bat * 4 + i2];
      float nm = sVec[w][O_PM + bo + li] - wgt * kin;
      sVec[w][O_MEAN + bo + li] = nm;
      if (li < 4) out[bb * rowStride + (long)t * 4 + li] = nm;
    }

    // Phase E: new_cov = (-K)_blk * (HP)_blk + P   (2 WMMAs, K-dim = 8)
#pragma unroll
    for (int k0 = 0; k0 < 2; ++k0) p = wmma4(aE[k0], bE[k0], p);
#pragma unroll
    for (int r = 0; r < 8; ++r) sCov[w][r + 8 * hi][li] = p[r];
  }
}

extern "C" void kernel_launch(void* const* d_in, const int* in_sizes, int n_in,
                              void* d_out, int out_size, void* d_ws, size_t ws_size,
                              hipStream_t stream) {
  (void)d_ws; (void)ws_size; (void)n_in; (void)out_size;
  const int T = 512;                              // reference shape
  const int B = in_sizes[0] / (T * 4);            // measurements: (B,T,4)
  const float* meas = (const float*)d_in[0];
  const float* Fd   = (const float*)d_in[1];
  const float* Fv   = (const float*)d_in[2];
  const float* Hb   = (const float*)d_in[3];
  const float* qs   = (const float*)d_in[4];
  const float* rs   = (const float*)d_in[5];
  const float* mw   = (const float*)d_in[6];
  const float* W1   = (const float*)d_in[7];
  const float* b1   = (const float*)d_in[8];
  const float* g1   = (const float*)d_in[9];
  const float* be1  = (const float*)d_in[10];
  const float* W2   = (const float*)d_in[11];
  const float* b2   = (const float*)d_in[12];
  float* out = (float*)d_out;

  int waves  = (B + 1) / 2;                       // 2 batches per wave
  int blocks = (waves + WPB - 1) / WPB;
  kalman_scan_kernel<<<blocks, WPB * 32, 0, stream>>>(
      meas, Fd, Fv, Hb, qs, rs, mw, W1, b1, g1, be1, W2, b2, out, B, T);
}